// FunctorF_v2_71262097375899
// MI455X (gfx1250) — compile-verified
//
#include <hip/hip_runtime.h>
#include <hip/hip_bf16.h>

typedef __attribute__((ext_vector_type(2))) float v2f;
typedef __attribute__((ext_vector_type(8))) float v8f;

#define NPTS   4096
#define NBATCH 8
#define KNN    16
#define HDIM   128
#define GDIM   16
#define ADIM   32
#define H1DIM  64

#define H1_STRIDE 68   // 68 % 64 == 4 -> bank-conflict-free row stride
#define F_STRIDE  132  // 132 % 64 == 4

// ---------------------------------------------------------------------------
// Kernel A: k-NN (K=16, Gram distances) + local context -> X (B*N x 6).
// 256 threads / block, 2 threads per query (each scans half the candidates),
// per-half sorted top-16 lists merged via LDS with stable tie ordering.
// The 48 KB LDS position cache is reused as the merge buffer after a barrier.
// ---------------------------------------------------------------------------
__global__ __launch_bounds__(256) void knn_context_kernel(
    const float* __restrict__ pos, float* __restrict__ X) {
  __shared__ float smem[NPTS * 3];  // 48 KB: positions, then merge lists
  const int b    = blockIdx.y;
  const int tid  = threadIdx.x;
  const int q    = blockIdx.x * 128 + (tid & 127);
  const int half = tid >> 7;                      // 0: j in [0,2048), 1: [2048,4096)
  const float* posb = pos + (size_t)b * NPTS * 3;

  for (int t = tid; t < NPTS * 3; t += 256) smem[t] = posb[t];
  __syncthreads();

  const float px = smem[q * 3 + 0], py = smem[q * 3 + 1], pz = smem[q * 3 + 2];
  const float sqi = px * px + py * py + pz * pz;

  float bd[KNN];
  int   bi[KNN];
#pragma unroll
  for (int k = 0; k < KNN; ++k) { bd[k] = 3.4e38f; bi[k] = 0; }

  const int j0 = half * (NPTS / 2);
  for (int j = j0; j < j0 + NPTS / 2; ++j) {
    const float qx = smem[j * 3 + 0], qy = smem[j * 3 + 1], qz = smem[j * 3 + 2];
    const float sqj = qx * qx + qy * qy + qz * qz;
    const float dot = px * qx + py * qy + pz * qz;
    float d2 = sqi + sqj - 2.0f * dot;            // same Gram form as reference
    d2 = fmaxf(d2, 0.0f);
    if (d2 < bd[KNN - 1]) {                       // stable insert (ties keep low j)
      float dd = d2; int ii = j;
#pragma unroll
      for (int k = 0; k < KNN; ++k) {
        if (dd < bd[k]) {
          float tf = bd[k]; bd[k] = dd; dd = tf;
          int   ti = bi[k]; bi[k] = ii; ii = ti;
        }
      }
    }
  }
  __syncthreads();                                // done reading positions

  // Reuse smem: 256*16 dists (16 KB) + 256*16 indices (16 KB) = 32 KB < 48 KB
  float* md = smem;
  int*   mi = (int*)smem + 256 * KNN;
#pragma unroll
  for (int k = 0; k < KNN; ++k) {
    md[tid * KNN + k] = bd[k];
    mi[tid * KNN + k] = bi[k];
  }
  __syncthreads();

  if (tid < 128) {                                // one merger per query
    const float* da = md + tid * KNN;             // low-index half (j < 2048)
    const float* db = md + (tid + 128) * KNN;     // high-index half
    const int*   ja = mi + tid * KNN;
    const int*   jb = mi + (tid + 128) * KNN;
    int ia = 0, ib = 0;
    float sx = 0.f, sy = 0.f, sz = 0.f;
#pragma unroll
    for (int k = 0; k < KNN; ++k) {               // two-pointer merge, 16 smallest
      const float va = da[ia], vb = db[ib];
      const bool takeA = va <= vb;                // tie -> lower j (matches top_k)
      const int  j = takeA ? ja[ia] : jb[ib];
      ia += takeA ? 1 : 0;
      ib += takeA ? 0 : 1;
      sx += posb[j * 3 + 0];                      // gather from global (L2-resident)
      sy += posb[j * 3 + 1];
      sz += posb[j * 3 + 2];
    }
    const float inv = 1.0f / (float)KNN;
    float* xr = X + ((size_t)b * NPTS + q) * 6;
    xr[0] = px; xr[1] = py; xr[2] = pz;
    xr[3] = sx * inv - px; xr[4] = sy * inv - py; xr[5] = sz * inv - pz;
  }
}

// ---------------------------------------------------------------------------
// Kernel C: FiLM parameters  gamma = goal@Wg + bg, beta = goal@Wb + bb
// ---------------------------------------------------------------------------
__global__ __launch_bounds__(HDIM) void film_kernel(
    const float* __restrict__ goal,
    const float* __restrict__ Wg, const float* __restrict__ bg,
    const float* __restrict__ Wb, const float* __restrict__ bb,
    float* __restrict__ gamma, float* __restrict__ beta) {
  const int b = blockIdx.x;
  const int h = threadIdx.x;
  float g = bg[h], e = bb[h];
  for (int t = 0; t < GDIM; ++t) {
    const float gv = goal[b * GDIM + t];
    g += gv * Wg[t * HDIM + h];
    e += gv * Wb[t * HDIM + h];
  }
  gamma[b * HDIM + h] = g;
  beta[b * HDIM + h]  = e;
}

// ---------------------------------------------------------------------------
// Kernel B: fused MLP with V_WMMA_F32_16X16X4_F32 (full fp32 matrix cores).
// Block = 128 threads = 4 waves; each wave owns a 16-row tile.
// Two independent accumulator chains per K step share one A-fragment.
// ---------------------------------------------------------------------------
__global__ __launch_bounds__(128) void mlp_wmma_kernel(
    const float* __restrict__ X,
    const float* __restrict__ W1, const float* __restrict__ b1,
    const float* __restrict__ W2, const float* __restrict__ b2,
    const float* __restrict__ Wa, const float* __restrict__ ba,
    const float* __restrict__ gamma, const float* __restrict__ beta,
    float* __restrict__ out) {
  __shared__ float h1s[4][16 * H1_STRIDE];  // per-wave 16x64 activations
  __shared__ float fs[4][16 * F_STRIDE];    // per-wave 16x128 activations

  const int lane  = threadIdx.x & 31;
  const int wave  = threadIdx.x >> 5;
  const int row0  = blockIdx.x * 64 + wave * 16;
  const int batch = row0 >> 12;             // 4096 rows per batch
  const int m     = lane & 15;
  const bool hi   = lane >= 16;             // hi lanes carry K+2, K+3
  const int  koff = hi ? 2 : 0;
  const float kmask = hi ? 0.0f : 1.0f;     // branchless K-padding (no saveexec)

  // ---------------- layer 1: X(16x6, pad K->8) @ W1(6x64) ----------------
  const float* xr = X + (size_t)(row0 + m) * 6;
  v2f a0, a1;
  a0.x = xr[koff];
  a0.y = xr[koff + 1];
  a1.x = xr[4] * kmask;                     // unconditional load + mask
  a1.y = xr[5] * kmask;

#pragma unroll
  for (int nt = 0; nt < 4; ++nt) {
    const int col = nt * 16 + m;
    v2f bv0, bv1;
    bv0.x = W1[koff * H1DIM + col];
    bv0.y = W1[(koff + 1) * H1DIM + col];
    bv1.x = W1[4 * H1DIM + col] * kmask;
    bv1.y = W1[5 * H1DIM + col] * kmask;
    v8f acc = {};
    acc = __builtin_amdgcn_wmma_f32_16x16x4_f32(false, a0, false, bv0,
                                                (short)0, acc, false, false);
    acc = __builtin_amdgcn_wmma_f32_16x16x4_f32(false, a1, false, bv1,
                                                (short)0, acc, false, false);
    const float bias = b1[col];
#pragma unroll
    for (int r = 0; r < 8; ++r) {
      const int mr = hi ? (r + 8) : r;
      h1s[wave][mr * H1_STRIDE + col] = fmaxf(acc[r] + bias, 0.0f);
    }
  }
  __syncthreads();

  // ------------- layer 2: h1(16x64) @ W2(64x128), ReLU, FiLM --------------
#pragma unroll
  for (int ntp = 0; ntp < 4; ++ntp) {       // two 16-col tiles per pass
    const int col0 = ntp * 32 + m;
    const int col1 = col0 + 16;
    v8f acc0 = {}, acc1 = {};
#pragma unroll 4
    for (int kk = 0; kk < H1DIM; kk += 4) {
      v2f a;                                 // shared A-fragment
      a.x = h1s[wave][m * H1_STRIDE + kk + koff];
      a.y = h1s[wave][m * H1_STRIDE + kk + koff + 1];
      v2f bA, bB;
      bA.x = W2[(kk + koff) * HDIM + col0];
      bA.y = W2[(kk + koff + 1) * HDIM + col0];
      bB.x = W2[(kk + koff) * HDIM + col1];
      bB.y = W2[(kk + koff + 1) * HDIM + col1];
      acc0 = __builtin_amdgcn_wmma_f32_16x16x4_f32(false, a, false, bA,
                                                   (short)0, acc0, false, false);
      acc1 = __builtin_amdgcn_wmma_f32_16x16x4_f32(false, a, false, bB,
                                                   (short)0, acc1, false, false);
    }
    const float bias0 = b2[col0], bias1 = b2[col1];
    const float g0 = gamma[batch * HDIM + col0], g1 = gamma[batch * HDIM + col1];
    const float e0 = beta[batch * HDIM + col0],  e1 = beta[batch * HDIM + col1];
#pragma unroll
    for (int r = 0; r < 8; ++r) {
      const int mr = hi ? (r + 8) : r;
      const float f0 = fmaxf(acc0[r] + bias0, 0.0f);
      const float f1 = fmaxf(acc1[r] + bias1, 0.0f);
      fs[wave][mr * F_STRIDE + col0] = g0 * f0 + e0;
      fs[wave][mr * F_STRIDE + col1] = g1 * f1 + e1;
    }
  }
  __syncthreads();

  // ------------- layer 3: f(16x128) @ Wa(128x32), ReLU -> out -------------
  {
    const int col0 = m;
    const int col1 = 16 + m;
    v8f acc0 = {}, acc1 = {};
#pragma unroll 4
    for (int kk = 0; kk < HDIM; kk += 4) {
      v2f a;
      a.x = fs[wave][m * F_STRIDE + kk + koff];
      a.y = fs[wave][m * F_STRIDE + kk + koff + 1];
      v2f bA, bB;
      bA.x = Wa[(kk + koff) * ADIM + col0];
      bA.y = Wa[(kk + koff + 1) * ADIM + col0];
      bB.x = Wa[(kk + koff) * ADIM + col1];
      bB.y = Wa[(kk + koff + 1) * ADIM + col1];
      acc0 = __builtin_amdgcn_wmma_f32_16x16x4_f32(false, a, false, bA,
                                                   (short)0, acc0, false, false);
      acc1 = __builtin_amdgcn_wmma_f32_16x16x4_f32(false, a, false, bB,
                                                   (short)0, acc1, false, false);
    }
    const float bias0 = ba[col0], bias1 = ba[col1];
#pragma unroll
    for (int r = 0; r < 8; ++r) {
      const int mr = hi ? (r + 8) : r;
      out[(size_t)(row0 + mr) * ADIM + col0] = fmaxf(acc0[r] + bias0, 0.0f);
      out[(size_t)(row0 + mr) * ADIM + col1] = fmaxf(acc1[r] + bias1, 0.0f);
    }
  }
}

// ---------------------------------------------------------------------------
extern "C" void kernel_launch(void* const* d_in, const int* in_sizes, int n_in,
                              void* d_out, int out_size, void* d_ws, size_t ws_size,
                              hipStream_t stream) {
  const float* pos  = (const float*)d_in[0];
  const float* goal = (const float*)d_in[1];
  const float* W1   = (const float*)d_in[2];
  const float* b1   = (const float*)d_in[3];
  const float* W2   = (const float*)d_in[4];
  const float* b2   = (const float*)d_in[5];
  const float* Wg   = (const float*)d_in[6];
  const float* bg   = (const float*)d_in[7];
  const float* Wb   = (const float*)d_in[8];
  const float* bb   = (const float*)d_in[9];
  const float* Wa   = (const float*)d_in[10];
  const float* ba   = (const float*)d_in[11];

  float* Xf    = (float*)d_ws;                        // 32768*6 floats
  float* gamma = Xf + (size_t)NBATCH * NPTS * 6;      // 8*128
  float* beta  = gamma + NBATCH * HDIM;               // 8*128

  knn_context_kernel<<<dim3(NPTS / 128, NBATCH), 256, 0, stream>>>(pos, Xf);
  film_kernel<<<NBATCH, HDIM, 0, stream>>>(goal, Wg, bg, Wb, bb, gamma, beta);
  mlp_wmma_kernel<<<(NBATCH * NPTS) / 64, 128, 0, stream>>>(
      Xf, W1, b1, W2, b2, Wa, ba, gamma, beta, (float*)d_out);
}